// ObservedAggregateStgcn_18760417148964
// MI455X (gfx1250) — compile-verified
//
#include <hip/hip_runtime.h>

typedef __attribute__((ext_vector_type(2))) float v2f;
typedef __attribute__((ext_vector_type(8))) float v8f;

#define NP   3
#define NC   1024
#define NV   64
#define NF   512
#define NK   9
#define ND   63
#define HOUT 8            // NF - ND*(NK-1)
#define KDIM (NP * NV)    // 192

// -----------------------------------------------------------------------------
// Kernel 1: yT[c][j] = sum_{p,u} x[(p*NC+c)*NV + u] * graph[(p*NV+u)*NV + j]
// GEMM M=1024(c) x N=64(j) x K=192(p,u) using V_WMMA_F32_16X16X4_F32.
// One wave per 16x16 output tile; 48 K-steps of K=4.
//
// fp32 A 16x4 VGPR layout (ISA 7.12.2): lanes 0-15 row M=lane hold K=0(v0),K=1(v1);
// lanes 16-31 row M=lane-16 hold K=2(v0),K=3(v1). B 4x16 mirrored (K in lane
// halves, N striped across lanes). D: VGPR r = row (c0 + half*8 + r), col j0+lr.
// -----------------------------------------------------------------------------
__global__ __launch_bounds__(256) void stgcn_graph_wmma(
    const float* __restrict__ x, const float* __restrict__ graph,
    float* __restrict__ yT)
{
  const int wave = (blockIdx.x * blockDim.x + threadIdx.x) >> 5;  // 0..255
  const int lane = threadIdx.x & 31;
  const int half = lane >> 4;
  const int lr   = lane & 15;

  const int jT = wave & 3;    // 4 tiles along N (64/16)
  const int cT = wave >> 2;   // 64 tiles along M (1024/16)
  const int c0 = cT * 16;
  const int j0 = jT * 16;

  const int ca = c0 + lr;     // this lane's A row
  const int jb = j0 + lr;     // this lane's B/D column

  v8f acc = {};
#pragma unroll
  for (int kk = 0; kk < KDIM / 4; ++kk) {
    const int kg = kk * 4 + half * 2;   // lane's first global-K (even)
    const int p  = kg >> 6;
    const int u  = kg & 63;
    // A: x[(p*NC + ca)*NV + u], x[... + u+1]  -> contiguous, 8B aligned
    v2f a = *(const v2f*)(x + (size_t)(p * NC + ca) * NV + u);
    // B: graph[(p*NV + u)*NV + jb], graph[(p*NV + u + 1)*NV + jb]
    v2f b;
    b.x = graph[(size_t)(p * NV + u)     * NV + jb];
    b.y = graph[(size_t)(p * NV + u + 1) * NV + jb];
    acc = __builtin_amdgcn_wmma_f32_16x16x4_f32(
        /*neg_a=*/false, a, /*neg_b=*/false, b,
        /*c_mod=*/(short)0, acc, /*reuse_a=*/false, /*reuse_b=*/false);
  }

  const int rowBase = c0 + half * 8;
#pragma unroll
  for (int r = 0; r < 8; ++r) {
    yT[(size_t)(rowBase + r) * NV + jb] = acc[r];
  }
}

// -----------------------------------------------------------------------------
// Kernel 2: out[c,h,v] = (h==0 ? yT[c,v] : fifo[c,h-1,v])
//                        + sum_{i=1..8} fifo[c, i*63 + h - 1, v]
// One thread per 4 consecutive v (float4 coalesced); 131072 threads total.
// Pure HBM-streaming kernel: ~19 MB read, 2 MB write.
// -----------------------------------------------------------------------------
__global__ __launch_bounds__(256) void stgcn_fifo_sum(
    const float* __restrict__ fifo, const float* __restrict__ yT,
    float* __restrict__ out)
{
  const int t   = blockIdx.x * blockDim.x + threadIdx.x;  // 0..131071
  const int c   = t >> 7;          // / (HOUT * NV/4)
  const int rem = t & 127;
  const int h   = rem >> 4;        // 0..7
  const int v   = (rem & 15) << 2; // 0..60 step 4

  const float* fbase = fifo + (size_t)c * NF * NV + v;

  float4 acc;
  if (h == 0) {
    acc = *(const float4*)(yT + (size_t)c * NV + v);       // freshly shifted-in row
  } else {
    acc = *(const float4*)(fbase + (size_t)(h - 1) * NV);  // new_fifo[c,h,v]
  }
#pragma unroll
  for (int i = 1; i < NK; ++i) {
    float4 f = *(const float4*)(fbase + (size_t)(i * ND + h - 1) * NV);
    acc.x += f.x; acc.y += f.y; acc.z += f.z; acc.w += f.w;
  }
  *(float4*)(out + (size_t)t * 4) = acc;
}

// -----------------------------------------------------------------------------
extern "C" void kernel_launch(void* const* d_in, const int* in_sizes, int n_in,
                              void* d_out, int out_size, void* d_ws, size_t ws_size,
                              hipStream_t stream) {
  (void)in_sizes; (void)n_in; (void)out_size; (void)ws_size;
  const float* x     = (const float*)d_in[0];   // (1, P*C, 1, V) fp32
  const float* graph = (const float*)d_in[1];   // (P, V, V)      fp32
  const float* fifo  = (const float*)d_in[2];   // (1,1,C,F,V)    fp32
  float* out = (float*)d_out;                   // (1, C, HOUT, V)
  float* yT  = (float*)d_ws;                    // 1024*64 fp32 = 256 KB scratch

  // GEMM: 256 tiles, 8 waves/block -> 32 blocks of 256 threads
  stgcn_graph_wmma<<<32, 256, 0, stream>>>(x, graph, yT);
  // Dilated 9-tap sum: 131072 threads -> 512 blocks of 256
  stgcn_fifo_sum<<<512, 256, 0, stream>>>(fifo, yT, out);
}